// LlamaAttention_63093069578759
// MI455X (gfx1250) — compile-verified
//
#include <hip/hip_runtime.h>
#include <hip/hip_bf16.h>

// ---------------------------------------------------------------------------
// LlamaAttention decode layer for MI455X (gfx1250, wave32, WMMA).
// Memory-bound (~1 GB traffic vs ~10 GFLOP) => keep everything fp32 and use
// V_WMMA_F32_16X16X4_F32 for all matmuls (full precision, matrix pipe).
// Data movement (this round: fully pipelined):
//   - GEMMs: double-buffered GLOBAL_LOAD_ASYNC_TO_LDS_B128 (ASYNCcnt),
//     chunk i+1 streams into LDS while chunk i feeds the WMMAs.
//   - Attention: TENSOR_LOAD_TO_LDS (TDM) for K tiles, software-pipelined:
//     tile i+1's DMA overlaps tile i's softmax + P@V.
// ---------------------------------------------------------------------------

#define H_SIZE      4096
#define N_HEADS     32
#define N_KV        8
#define HEAD_DIM    128
#define GROUP       4                      // N_HEADS / N_KV
#define Q_SIZE      4096                   // N_HEADS * HEAD_DIM
#define KV_SIZE     1024                   // N_KV * HEAD_DIM
#define QKV_SIZE    (Q_SIZE + 2 * KV_SIZE) // 6144
#define NUM_SEQS    64
#define BLOCK_SZ    16
#define BLOCKS_PER_SEQ 128
#define ATT_SCALE   0.08838834764831845f   // HEAD_DIM^-0.5

typedef float v2f __attribute__((ext_vector_type(2)));
typedef float v8f __attribute__((ext_vector_type(8)));
typedef unsigned int v4u __attribute__((ext_vector_type(4)));
typedef int v4i __attribute__((ext_vector_type(4)));
typedef int v8i __attribute__((ext_vector_type(8)));

// LDS aperture: flat addr[31:0] == LDS byte offset (ISA 10.2), so truncating
// a generic pointer to a __shared__ object yields the DS address.
__device__ __forceinline__ unsigned lds_addr_of(const void* p) {
    return (unsigned)(unsigned long long)(uintptr_t)p;
}

// Async 16B global->LDS copy, per active lane, tracked by ASYNCcnt.
__device__ __forceinline__ void async_copy_b128(void* lds, const void* gsrc) {
    asm volatile("global_load_async_to_lds_b128 %0, %1, off"
                 :: "v"(lds_addr_of(lds)), "v"(gsrc) : "memory");
}
__device__ __forceinline__ void wait_async0() {
    asm volatile("s_wait_asynccnt 0x0" ::: "memory");
}
// ASYNCcnt is per-wave and completes in order; each GEMM chunk issues exactly
// 5 async instructions, so <=5 outstanding means the previous chunk landed.
__device__ __forceinline__ void wait_async5() {
    asm volatile("s_wait_asynccnt 0x5" ::: "memory");
}
__device__ __forceinline__ void wait_ds0() {
    asm volatile("s_wait_dscnt 0x0" ::: "memory");
}

// TDM: load a 16-row x 128-float tile (row stride 1024 floats) into LDS with
// 4-DWORD padding every 128 DWORDs => lands exactly in float[16][132].
// clang-23 toolchain form: 6 args (g0 v4u, g1 v8i, g2 v4i, g3 v4i, v8i, cpol).
__device__ __forceinline__ void tdm_load_k_tile(unsigned lds_off, const float* tile) {
    unsigned long long ga = (unsigned long long)(uintptr_t)tile;
    v4u g0;
    g0[0] = 1u;                                     // count=1, user descriptor
    g0[1] = lds_off;                                // lds_addr
    g0[2] = (unsigned)(ga & 0xffffffffu);           // global_addr[31:0]
    g0[3] = (unsigned)((ga >> 32) & 0x01ffffffu)    // global_addr[56:32]
          | 0x80000000u;                            // type=2 ("image")
    v8i g1;
    g1[0] = (int)((2u << 16)     // data_size = 4B
                | (1u << 20)     // pad_enable
                | (6u << 22)     // pad_interval: 128 DWORDs
                | (3u << 25));   // pad_amount:   4 DWORDs
    g1[1] = (int)(128u << 16);   // tensor_dim0 = 128 (elements)
    g1[2] = (int)(16u  << 16);   // tensor_dim1 = 16
    g1[3] = (int)(128u << 16);   // tile_dim0   = 128
    g1[4] = 16;                  // tile_dim1   = 16 (tile_dim2 = 0)
    g1[5] = 1024;                // tensor_dim0_stride = 1024 elements
    g1[6] = 0;
    g1[7] = 0;
    v4i gz; gz[0] = 0; gz[1] = 0; gz[2] = 0; gz[3] = 0;
    v8i gz8; gz8[0] = 0; gz8[1] = 0; gz8[2] = 0; gz8[3] = 0;
    gz8[4] = 0; gz8[5] = 0; gz8[6] = 0; gz8[7] = 0;
    asm volatile("" ::: "memory");   // keep prior LDS accesses ahead of the DMA
    __builtin_amdgcn_tensor_load_to_lds(g0, g1, gz, gz, gz8, 0);
}

// ===========================================================================
// GEMM: C[64,N] = A[64,K] @ B[N,K]^T   (A=activations, B=row-major weights)
// 128 threads = 4 waves, each wave owns a 16x16 tile of a 64x16 block tile.
// fp32 WMMA 16x16x4; double-buffered async K-chunk streaming.
// ===========================================================================
#define KC 32

__global__ __launch_bounds__(128)
void gemm64_wmma(const float* __restrict__ A, const float* __restrict__ B,
                 float* __restrict__ C, int N, int K) {
    __shared__ float Ash[2][64][KC + 4];   // +4 pad: conflict-free & 16B rows
    __shared__ float Bsh[2][16][KC + 4];

    const int tid  = threadIdx.x;
    const int wv   = tid >> 5;
    const int lane = tid & 31;
    const int lo   = lane & 15;
    const int hi   = lane >> 4;
    const int n0   = blockIdx.x * 16;
    const int m0   = wv * 16;

    // Issue one K-chunk (5 async b128 per thread: 4 for A, 1 for B).
    auto issue_chunk = [&](int buf, int k0) {
#pragma unroll
        for (int j = 0; j < 4; j++) {
            int f  = tid + 128 * j;
            int r  = f >> 3;
            int c4 = (f & 7) << 2;
            async_copy_b128(&Ash[buf][r][c4], &A[(size_t)r * K + k0 + c4]);
        }
        int r  = tid >> 3;
        int c4 = (tid & 7) << 2;
        async_copy_b128(&Bsh[buf][r][c4], &B[(size_t)(n0 + r) * K + k0 + c4]);
    };

    v8f acc;
#pragma unroll
    for (int i = 0; i < 8; i++) acc[i] = 0.0f;

    const int nchunks = K / KC;
    issue_chunk(0, 0);                                   // prologue

    for (int c = 0; c < nchunks; c++) {
        const int  cur  = c & 1;
        const bool more = (c + 1) < nchunks;
        if (more) issue_chunk(cur ^ 1, (c + 1) * KC);    // stream next chunk
        if (more) wait_async5(); else wait_async0();     // current chunk landed
        __syncthreads();

#pragma unroll
        for (int t = 0; t < KC / 4; t++) {
            int k = 4 * t + 2 * hi;      // A lanes 0-15: K+0/1, lanes 16-31: K+2/3
            v2f a, b;
            a[0] = Ash[cur][m0 + lo][k];
            a[1] = Ash[cur][m0 + lo][k + 1];
            b[0] = Bsh[cur][lo][k];      // B[k][n]: col n = lo, mirrored K split
            b[1] = Bsh[cur][lo][k + 1];
            acc = __builtin_amdgcn_wmma_f32_16x16x4_f32(
                false, a, false, b, (short)0, acc, false, false);
        }
        __syncthreads();                 // all waves done before buf reuse
    }

    // D layout: VGPR v, lanes 0-15 -> row m0+v, lanes 16-31 -> row m0+v+8
#pragma unroll
    for (int v = 0; v < 8; v++) {
        int m = m0 + v + 8 * hi;
        C[(size_t)m * N + n0 + lo] = acc[v];
    }
}

// ===========================================================================
// RoPE on q (folds in attention scale) and k. grid (64 seqs, 40 heads), 64 thr.
// ===========================================================================
__global__ __launch_bounds__(64)
void rope_scale_kernel(const float* __restrict__ qkv, const int* __restrict__ positions,
                       float* __restrict__ qr, float* __restrict__ kr) {
    const int s = blockIdx.x;
    const int h = blockIdx.y;      // 0..31 q heads, 32..39 k heads
    const int i = threadIdx.x;     // 0..63 (rotary pair index)

    float p    = (float)positions[s];
    float freq = powf(10000.0f, -(float)i * (1.0f / 64.0f));
    float ang  = p * freq;
    float cs   = cosf(ang);
    float sn   = sinf(ang);

    const float* src;
    float*       dst;
    float        sc;
    if (h < N_HEADS) {
        src = qkv + (size_t)s * QKV_SIZE + h * HEAD_DIM;
        dst = qr  + (size_t)s * Q_SIZE   + h * HEAD_DIM;
        sc  = ATT_SCALE;                       // pre-scale q
    } else {
        int kh = h - N_HEADS;
        src = qkv + (size_t)s * QKV_SIZE + Q_SIZE + kh * HEAD_DIM;
        dst = kr  + (size_t)s * KV_SIZE  + kh * HEAD_DIM;
        sc  = 1.0f;
    }
    float x1 = src[i];
    float x2 = src[i + 64];
    dst[i]      = (x1 * cs - x2 * sn) * sc;
    dst[i + 64] = (x2 * cs + x1 * sn) * sc;
}

// ===========================================================================
// Paged flash attention. grid (8 kv heads, 64 seqs), 128 threads = 4 waves.
// Each wave handles ctx tiles it = w, w+4, ... (one 16-token cache block per
// tile == one WMMA N=16 tile). K tiles DMA'd by the TDM, pipelined one tile
// ahead; online softmax per wave; LDS merge with the current token folded in.
// ===========================================================================
#define NW 4

__global__ __launch_bounds__(128)
void attn_kernel(const float* __restrict__ qr, const float* __restrict__ kr,
                 const float* __restrict__ qkv,
                 const float* __restrict__ k_cache, const float* __restrict__ v_cache,
                 const int* __restrict__ block_tables, const int* __restrict__ context_lens,
                 float* __restrict__ attn) {
    const int kv   = blockIdx.x;
    const int s    = blockIdx.y;
    const int tid  = threadIdx.x;
    const int w    = tid >> 5;
    const int lane = tid & 31;
    const int lo   = lane & 15;
    const int hi   = lane >> 4;

    __shared__ float Ksh[NW][16][HEAD_DIM + 4]; // pad 4: matches TDM pad config
    __shared__ float Psh[NW][16][20];           // P tile staged for A-fragment layout
    __shared__ float Osh[NW][GROUP][HEAD_DIM];  // per-wave partial outputs
    __shared__ float Msh[NW][GROUP];
    __shared__ float Lsh[NW][GROUP];
    __shared__ float Snew[GROUP];

    // Zero P tile once; rows 4..15 stay zero forever => padded query rows of
    // the 16x16 WMMA contribute exact zeros (no NaN leakage into valid rows).
    for (int i = lane; i < 16 * 20; i += 32) ((float*)Psh[w])[i] = 0.0f;

    // Wave-uniform LDS destination for the TDM descriptor (SGPR-resident D#).
    const unsigned kslab = __builtin_amdgcn_readfirstlane(lds_addr_of(&Ksh[w][0][0]));

    // q A-fragments, resident in VGPRs for the whole kernel (q is pre-scaled).
    const float* qbase = qr + (size_t)s * Q_SIZE + kv * GROUP * HEAD_DIM;
    float qa[64];
#pragma unroll
    for (int t = 0; t < 32; t++) {
        int d = 4 * t + 2 * hi;
        if (lo < GROUP) {
            qa[2 * t]     = qbase[lo * HEAD_DIM + d];
            qa[2 * t + 1] = qbase[lo * HEAD_DIM + d + 1];
        } else {
            qa[2 * t]     = 0.0f;
            qa[2 * t + 1] = 0.0f;
        }
    }

    v8f acc[8];
#pragma unroll
    for (int c = 0; c < 8; c++)
#pragma unroll
        for (int i = 0; i < 8; i++) acc[c][i] = 0.0f;

    float m_run[GROUP], l_run[GROUP];
#pragma unroll
    for (int g = 0; g < GROUP; g++) { m_run[g] = -1e30f; l_run[g] = 0.0f; }

    const int pos    = context_lens[s] - 1;        // cache entries: [0, pos)
    const int ntiles = (pos + 15) >> 4;

    // Prologue: kick off this wave's first K-tile DMA.
    if (w < ntiles) {
        const int blk0 = __builtin_amdgcn_readfirstlane(
                             block_tables[s * BLOCKS_PER_SEQ + w]);
        tdm_load_k_tile(kslab, k_cache + ((size_t)blk0 * BLOCK_SZ * N_KV + kv) * HEAD_DIM);
    }

    for (int it = w; it < ntiles; it += NW) {
        const int blk = __builtin_amdgcn_readfirstlane(
                            block_tables[s * BLOCKS_PER_SEQ + it]);
        const float* vb = v_cache + ((size_t)blk * BLOCK_SZ * N_KV + kv) * HEAD_DIM;

        // Wait for the in-flight K-tile DMA (wave-private slab, no barrier).
        __builtin_amdgcn_s_wait_tensorcnt(0);
        asm volatile("" ::: "memory");

        // S = q @ K^T : M=16(4 valid q rows), N=16 ctx, K=128 -> 32 WMMAs.
        v8f S;
#pragma unroll
        for (int i = 0; i < 8; i++) S[i] = 0.0f;
#pragma unroll
        for (int t = 0; t < HEAD_DIM / 4; t++) {
            int k = 4 * t + 2 * hi;
            v2f a, b;
            a[0] = qa[2 * t];
            a[1] = qa[2 * t + 1];
            b[0] = Ksh[w][lo][k];        // B[k=dim][n=tok]: K[tok=lo][dim]
            b[1] = Ksh[w][lo][k + 1];
            S = __builtin_amdgcn_wmma_f32_16x16x4_f32(
                false, a, false, b, (short)0, S, false, false);
        }

        // K tile consumed: retire its LDS reads, then pipeline the next tile's
        // DMA so it overlaps softmax + P@V below. Also prefetch next V rows.
        const int itn = it + NW;
        if (itn < ntiles) {
            const int bn = __builtin_amdgcn_readfirstlane(
                               block_tables[s * BLOCKS_PER_SEQ + itn]);
            const float* kbn = k_cache + ((size_t)bn * BLOCK_SZ * N_KV + kv) * HEAD_DIM;
            const float* vbn = v_cache + ((size_t)bn * BLOCK_SZ * N_KV + kv) * HEAD_DIM;
            wait_ds0();                  // Ksh ds_loads done before DMA overwrite
            tdm_load_k_tile(kslab, kbn);
            size_t off = (size_t)lo * (N_KV * HEAD_DIM * 4) + (size_t)hi * 256;
            __builtin_prefetch((const char*)vbn + off, 0, 0);
        }

        // Online softmax per query row g (row g lives in VGPR g, ctx = lane).
        const int  nbase = it * 16;
        const bool valid = (nbase + lo) < pos;
#pragma unroll
        for (int g = 0; g < GROUP; g++) {
            float sv = valid ? S[g] : -1e30f;
            float m  = sv;
            m = fmaxf(m, __shfl_xor(m, 1));
            m = fmaxf(m, __shfl_xor(m, 2));
            m = fmaxf(m, __shfl_xor(m, 4));
            m = fmaxf(m, __shfl_xor(m, 8));
            float mn = fmaxf(m_run[g], m);
            float p  = valid ? expf(sv - mn) : 0.0f;
            float r  = expf(m_run[g] - mn);
            float ps = p;
            ps += __shfl_xor(ps, 1);
            ps += __shfl_xor(ps, 2);
            ps += __shfl_xor(ps, 4);
            ps += __shfl_xor(ps, 8);
            l_run[g] = l_run[g] * r + ps;
            m_run[g] = mn;
#pragma unroll
            for (int c = 0; c < 8; c++) acc[c][g] *= r;
            if (lane < 16) Psh[w][g][lo] = p;   // lanes 0-15 hold valid rows
        }

        // acc += P @ V : A = P tile from LDS, B = V read straight from global
        // (each V element touched exactly once per workgroup => same DRAM
        //  traffic as LDS staging, fewer LDS bytes).
#pragma unroll
        for (int c = 0; c < 8; c++) {
            int d0 = c * 16;
#pragma unroll
            for (int kk = 0; kk < 4; kk++) {
                int tok = 4 * kk + 2 * hi;
                v2f a, b;
                a[0] = Psh[w][lo][4 * kk + 2 * hi];
                a[1] = Psh[w][lo][4 * kk + 2 * hi + 1];
                b[0] = vb[(size_t)tok * (N_KV * HEAD_DIM) + d0 + lo];
                b[1] = vb[(size_t)(tok + 1) * (N_KV * HEAD_DIM) + d0 + lo];
                acc[c] = __builtin_amdgcn_wmma_f32_16x16x4_f32(
                    false, a, false, b, (short)0, acc[c], false, false);
            }
        }
    }

    // Per-wave partials -> LDS.
    if (lane < 16) {
#pragma unroll
        for (int g = 0; g < GROUP; g++) {
#pragma unroll
            for (int c = 0; c < 8; c++) Osh[w][g][c * 16 + lo] = acc[c][g];
            if (lane == 0) { Msh[w][g] = m_run[g]; Lsh[w][g] = l_run[g]; }
        }
    }
    __syncthreads();

    // Current-token score s_new[g] = (scaled q_g) . k_new.
    if (tid < GROUP) {
        const float* kn = kr + (size_t)s * KV_SIZE + kv * HEAD_DIM;
        float a = 0.0f;
        for (int d = 0; d < HEAD_DIM; d++) a += qbase[tid * HEAD_DIM + d] * kn[d];
        Snew[tid] = a;
    }
    __syncthreads();

    // Flash merge across the 4 waves + the current token. 128 threads, d = tid.
    {
        const int d = tid;
        const float* vnew = qkv + (size_t)s * QKV_SIZE + Q_SIZE + KV_SIZE + kv * HEAD_DIM;
        float vn = vnew[d];
#pragma unroll
        for (int g = 0; g < GROUP; g++) {
            float M = Snew[g];
            for (int ww = 0; ww < NW; ww++) M = fmaxf(M, Msh[ww][g]);
            float e   = expf(Snew[g] - M);
            float num = e * vn;
            float den = e;
            for (int ww = 0; ww < NW; ww++) {
                float f = expf(Msh[ww][g] - M);
                num += f * Osh[ww][g][d];
                den += f * Lsh[ww][g];
            }
            attn[(size_t)s * Q_SIZE + (kv * GROUP + g) * HEAD_DIM + d] = num / den;
        }
    }
}

// ===========================================================================
// Host launch
// ===========================================================================
extern "C" void kernel_launch(void* const* d_in, const int* in_sizes, int n_in,
                              void* d_out, int out_size, void* d_ws, size_t ws_size,
                              hipStream_t stream) {
    const int*   positions    = (const int*)  d_in[0];
    const float* hidden       = (const float*)d_in[1];
    const float* k_cache      = (const float*)d_in[2];
    const float* v_cache      = (const float*)d_in[3];
    const int*   block_tables = (const int*)  d_in[4];
    const int*   context_lens = (const int*)  d_in[5];
    const float* W_qkv        = (const float*)d_in[6];
    const float* W_o          = (const float*)d_in[7];
    float*       out          = (float*)d_out;

    float* ws   = (float*)d_ws;
    float* qkv  = ws;                                  // 64*6144
    float* qr   = qkv + (size_t)NUM_SEQS * QKV_SIZE;   // 64*4096 (pre-scaled, roped)
    float* kr   = qr  + (size_t)NUM_SEQS * Q_SIZE;     // 64*1024 (roped)
    float* attn = kr  + (size_t)NUM_SEQS * KV_SIZE;    // 64*4096

    gemm64_wmma<<<dim3(QKV_SIZE / 16), 128, 0, stream>>>(hidden, W_qkv, qkv, QKV_SIZE, H_SIZE);
    rope_scale_kernel<<<dim3(NUM_SEQS, N_HEADS + N_KV), 64, 0, stream>>>(qkv, positions, qr, kr);
    attn_kernel<<<dim3(N_KV, NUM_SEQS), 128, 0, stream>>>(qr, kr, qkv, k_cache, v_cache,
                                                          block_tables, context_lens, attn);
    gemm64_wmma<<<dim3(H_SIZE / 16), 128, 0, stream>>>(attn, W_o, out, H_SIZE, Q_SIZE);
}